// RSAGE_Hetero_60284160966706
// MI455X (gfx1250) — compile-verified
//
#include <hip/hip_runtime.h>

typedef float v2f __attribute__((ext_vector_type(2)));
typedef float v8f __attribute__((ext_vector_type(8)));

#define NP 50000
#define NA 20000
#define DH 128
#define NC 40
#define EW 400000
#define EC 800000
#define ER 400000

// ---------------- small helpers ----------------

__global__ void k_izero(int* __restrict__ p, int n) {
  int i = blockIdx.x * blockDim.x + threadIdx.x;
  if (i < n) p[i] = 0;
}

__global__ void k_icount(const int* __restrict__ dst, int* __restrict__ cnt, int E) {
  int i = blockIdx.x * blockDim.x + threadIdx.x;
  if (i < E) atomicAdd(&cnt[dst[i]], 1);
}

__global__ void k_invcnt(const int* __restrict__ cnt, float* __restrict__ inv, int n) {
  int i = blockIdx.x * blockDim.x + threadIdx.x;
  if (i < n) inv[i] = 1.0f / (float)max(cnt[i], 1);
}

// o = 0.5*(a+b)   (combined root weights)
__global__ void k_halfsum(const float* __restrict__ a, const float* __restrict__ b,
                          float* __restrict__ o, int n) {
  int i = blockIdx.x * blockDim.x + threadIdx.x;
  if (i < n) o[i] = 0.5f * (a[i] + b[i]);
}

// o = 0.5*(a + b + ba[j] + bb[j])  (paper-dst relation average + biases)
__global__ void k_combine2(const float* __restrict__ a, const float* __restrict__ b,
                           const float* __restrict__ ba, const float* __restrict__ bb,
                           float* __restrict__ o, long rows, int D) {
  long i = (long)blockIdx.x * blockDim.x + threadIdx.x;
  if (i >= rows * (long)D) return;
  int j = (int)(i % D);
  o[i] = 0.5f * (a[i] + b[i] + ba[j] + bb[j]);
}

__global__ void k_addbias(float* __restrict__ a, const float* __restrict__ b,
                          long rows, int D) {
  long i = (long)blockIdx.x * blockDim.x + threadIdx.x;
  if (i >= rows * (long)D) return;
  a[i] += b[(int)(i % D)];
}

// ---------------- CSR build ----------------

// Single-block exclusive scan, 4 elements/thread per 4096-chunk.
__global__ __launch_bounds__(1024)
void k_scan(const int* __restrict__ cnt, int* __restrict__ off, int n) {
  __shared__ int sh[1024];
  __shared__ int carry;
  if (threadIdx.x == 0) carry = 0;
  __syncthreads();
  for (int base = 0; base < n; base += 4096) {
    int v[4];
    int i0 = base + threadIdx.x * 4;
    int tsum = 0;
#pragma unroll
    for (int j = 0; j < 4; ++j) {
      int i = i0 + j;
      v[j] = (i < n) ? cnt[i] : 0;
      tsum += v[j];
    }
    sh[threadIdx.x] = tsum;
    __syncthreads();
    for (int s = 1; s < 1024; s <<= 1) {
      int t = (threadIdx.x >= (unsigned)s) ? sh[threadIdx.x - s] : 0;
      __syncthreads();
      sh[threadIdx.x] += t;
      __syncthreads();
    }
    int excl = carry + sh[threadIdx.x] - tsum;
#pragma unroll
    for (int j = 0; j < 4; ++j) {
      int i = i0 + j;
      if (i < n) off[i] = excl;
      excl += v[j];
    }
    __syncthreads();
    if (threadIdx.x == 1023) carry += sh[1023];
    __syncthreads();
  }
  if (threadIdx.x == 0) off[n] = carry;
}

__global__ void k_fill(const int* __restrict__ ei, const int* __restrict__ off,
                       int* __restrict__ cursor, int* __restrict__ perm, int E) {
  int e = blockIdx.x * blockDim.x + threadIdx.x;
  if (e >= E) return;
  int src = ei[e];          // ei is [2,E] row-major
  int dst = ei[E + e];
  int p = atomicAdd(&cursor[dst], 1);
  perm[off[dst] + p] = src;
}

// ---------------- atomic-free mean aggregation (CSR gather-reduce) ----------------
// One block per destination row; neighbor ids staged in LDS; thread = feature col.

__global__ void k_gather(const float* __restrict__ feat, const int* __restrict__ off,
                         const int* __restrict__ perm, const float* __restrict__ inv,
                         float* __restrict__ outp, int D) {
  __shared__ int ids[256];
  const int dst = blockIdx.x;
  const int f = threadIdx.x;
  const int s = off[dst], e = off[dst + 1];
  float sum = 0.0f;
  for (int base = s; base < e; base += 256) {
    int m = min(256, e - base);
    for (int j = threadIdx.x; j < m; j += blockDim.x) ids[j] = perm[base + j];
    __syncthreads();
    if (f < D)
      for (int j = 0; j < m; ++j) sum += feat[(long)ids[j] * D + f];
    __syncthreads();
  }
  if (f < D) outp[(long)dst * D + f] = sum * inv[dst];
}

// ---------------- WMMA f32 GEMM:  C[M,N] (+)= A[M,128] @ W[128,N]  ----------------
// One wave per 16x16 tile; V_WMMA_F32_16X16X4_F32, K stepped by 4.
// N is a template constant so all A- and B-load offsets fold into 24-bit IOFFSETs.
// OOB lanes read clamped (valid) addresses: garbage pollutes only masked rows/cols.
// flags: bit0 = accumulate into C, bit1 = ReLU epilogue.

template <int N>
__global__ __launch_bounds__(128)
void k_gemm(const float* __restrict__ A, const float* __restrict__ W,
            float* __restrict__ C, int M, int flags) {
  const int K = 128;
  const int wave = threadIdx.x >> 5;
  const int lane = threadIdx.x & 31;
  constexpr int tiles_n = (N + 15) >> 4;
  const long tileId = (long)blockIdx.x * 4 + wave;
  const int tm = (int)(tileId / tiles_n);
  const int tn = (int)(tileId - (long)tm * tiles_n);
  if (tm * 16 >= M) return;                     // wave-uniform: EXEC stays full

  const int half = lane >> 4;                   // K pair (k,k+1) vs (k+2,k+3)
  const int l15  = lane & 15;
  const int rowA = tm * 16 + l15;               // A: M index per lane
  const int colB = tn * 16 + l15;               // B: N index per lane
  const bool colOK = colB < N;

  const float* aptr = A + (long)(rowA < M ? rowA : 0) * K + 2 * half;
  const float* bptr = W + (long)(2 * half) * N + (colOK ? colB : 0);

  v8f acc = {0.f, 0.f, 0.f, 0.f, 0.f, 0.f, 0.f, 0.f};
#pragma unroll
  for (int k = 0; k < K; k += 4) {
    v2f a = *(const v2f*)(aptr + k);            // A[row, k+2h : k+2h+1]
    v2f b;
    b[0] = bptr[k * N];                         // W[k+2h,   col]  (const offset)
    b[1] = bptr[(k + 1) * N];                   // W[k+2h+1, col]  (const offset)
    acc = __builtin_amdgcn_wmma_f32_16x16x4_f32(false, a, false, b,
                                                (short)0, acc, false, false);
  }

  // D layout: VGPR v -> row tm*16 + half*8 + v, col = tn*16 + l15
  if (colOK) {
    const int r0 = tm * 16 + half * 8;
#pragma unroll
    for (int v = 0; v < 8; ++v) {
      int r = r0 + v;
      if (r < M) {
        long idx = (long)r * N + colB;
        float val = acc[v];
        if (flags & 1) val += C[idx];
        if (flags & 2) val = fmaxf(val, 0.0f);
        C[idx] = val;
      }
    }
  }
}

// ---------------- launcher ----------------

static inline int cdiv(long a, long b) { return (int)((a + b - 1) / b); }

extern "C" void kernel_launch(void* const* d_in, const int* in_sizes, int n_in,
                              void* d_out, int out_size, void* d_ws, size_t ws_size,
                              hipStream_t stream) {
  (void)in_sizes; (void)n_in; (void)out_size; (void)ws_size;

  const float* x_paper  = (const float*)d_in[0];
  const float* x_author = (const float*)d_in[1];
  const int*   ei_w     = (const int*)d_in[2];
  const int*   ei_c     = (const int*)d_in[3];
  const int*   ei_r     = (const int*)d_in[4];
  const float *Wl0_w=(const float*)d_in[5],  *bl0_w=(const float*)d_in[6],  *Wr0_w=(const float*)d_in[7];
  const float *Wl0_c=(const float*)d_in[8],  *bl0_c=(const float*)d_in[9],  *Wr0_c=(const float*)d_in[10];
  const float *Wl0_r=(const float*)d_in[11], *bl0_r=(const float*)d_in[12], *Wr0_r=(const float*)d_in[13];
  const float *Wl1_w=(const float*)d_in[14], *bl1_w=(const float*)d_in[15], *Wr1_w=(const float*)d_in[16];
  const float *Wl1_c=(const float*)d_in[17], *bl1_c=(const float*)d_in[18], *Wr1_c=(const float*)d_in[19];
  float* out = (float*)d_out;
  float* ws  = (float*)d_ws;

  const long A128 = (long)NA * DH, P128 = (long)NP * DH, P40 = (long)NP * NC;
  long o = 0;
  float* T_a  = ws + o; o += A128;   // x_author @ Wl0_w   (layer1: U_a [NA,40])
  float* T_pc = ws + o; o += P128;   // x_paper  @ Wl0_c   (layer1: U_pc [NP,40])
  float* T_pr = ws + o; o += P128;   // x_paper  @ Wl0_r
  float* ACCW = ws + o; o += P128;   // writes agg -> becomes hp
  float* ACCC = ws + o; o += P128;   // cites agg  (layer1: A2W | A2C packed)
  float* ACCR = ws + o; o += A128;   // rev agg    -> becomes ha
  float* INVW = ws + o; o += NP;
  float* INVC = ws + o; o += NP;
  float* INVR = ws + o; o += NA;
  float* WC0  = ws + o; o += 128 * 128;  // 0.5*(Wr0_w+Wr0_c)
  float* WC1  = ws + o; o += 128 * 40;   // 0.5*(Wr1_w+Wr1_c)

  int* ib = (int*)(ws + o);
  long io = 0;
  int* CNT   = ib + io; io += NP;        // histogram, then reused as fill cursor
  int* OFFW  = ib + io; io += NP + 1;
  int* OFFC  = ib + io; io += NP + 1;
  int* OFFR  = ib + io; io += NA + 1;
  int* PERMW = ib + io; io += EW;
  int* PERMC = ib + io; io += EC;
  int* PERMR = ib + io; io += ER;

  auto gemm128 = [&](const float* A, const float* Wm, float* C, int M, int flags) {
    long tiles = (long)cdiv(M, 16) * 8;
    k_gemm<128><<<cdiv(tiles, 4), 128, 0, stream>>>(A, Wm, C, M, flags);
  };
  auto gemm40 = [&](const float* A, const float* Wm, float* C, int M, int flags) {
    long tiles = (long)cdiv(M, 16) * 3;
    k_gemm<40><<<cdiv(tiles, 4), 128, 0, stream>>>(A, Wm, C, M, flags);
  };

  // CSR per relation: histogram -> exclusive scan -> inverse degree -> fill
  auto build_csr = [&](const int* ei, int E, int n, int* off, int* perm, float* inv) {
    k_izero<<<cdiv(n, 256), 256, 0, stream>>>(CNT, n);
    k_icount<<<cdiv(E, 256), 256, 0, stream>>>(ei + E, CNT, E);
    k_scan<<<1, 1024, 0, stream>>>(CNT, off, n);
    k_invcnt<<<cdiv(n, 256), 256, 0, stream>>>(CNT, inv, n);
    k_izero<<<cdiv(n, 256), 256, 0, stream>>>(CNT, n);
    k_fill<<<cdiv(E, 256), 256, 0, stream>>>(ei, off, CNT, perm, E);
  };

  // ---- setup: combined root weights + CSR (edge structure is layer-invariant) ----
  k_halfsum<<<cdiv(128 * 128, 256), 256, 0, stream>>>(Wr0_w, Wr0_c, WC0, 128 * 128);
  k_halfsum<<<cdiv(128 * 40, 256), 256, 0, stream>>>(Wr1_w, Wr1_c, WC1, 128 * 40);
  build_csr(ei_w, EW, NP, OFFW, PERMW, INVW);
  build_csr(ei_c, EC, NP, OFFC, PERMC, INVC);
  build_csr(ei_r, ER, NA, OFFR, PERMR, INVR);

  // ---- layer 0: transform-first, then atomic-free mean gather ----
  gemm128(x_author, Wl0_w, T_a,  NA, 0);
  gemm128(x_paper,  Wl0_c, T_pc, NP, 0);
  gemm128(x_paper,  Wl0_r, T_pr, NP, 0);
  k_gather<<<NP, 128, 0, stream>>>(T_a,  OFFW, PERMW, INVW, ACCW, DH);
  k_gather<<<NP, 128, 0, stream>>>(T_pc, OFFC, PERMC, INVC, ACCC, DH);
  k_gather<<<NA, 128, 0, stream>>>(T_pr, OFFR, PERMR, INVR, ACCR, DH);
  // hp = relu(0.5*(aggW + aggC + bw + bc) + x_paper @ WC0)   (in place in ACCW)
  k_combine2<<<cdiv(P128, 256), 256, 0, stream>>>(ACCW, ACCC, bl0_w, bl0_c, ACCW, NP, DH);
  gemm128(x_paper, WC0, ACCW, NP, 1 | 2);
  // ha = relu(aggR + br + x_author @ Wr0_r)                  (in place in ACCR)
  k_addbias<<<cdiv(A128, 256), 256, 0, stream>>>(ACCR, bl0_r, NA, DH);
  gemm128(x_author, Wr0_r, ACCR, NA, 1 | 2);

  // ---- layer 1 (author output is dead code; only paper path) ----
  float* hp = ACCW;
  float* ha = ACCR;
  float* U_a  = T_a;           // [NA,40]
  float* U_pc = T_pc;          // [NP,40]
  float* A2W  = ACCC;          // [NP,40]
  float* A2C  = ACCC + P40;    // [NP,40]
  gemm40(ha, Wl1_w, U_a,  NA, 0);
  gemm40(hp, Wl1_c, U_pc, NP, 0);
  k_gather<<<NP, 64, 0, stream>>>(U_a,  OFFW, PERMW, INVW, A2W, NC);
  k_gather<<<NP, 64, 0, stream>>>(U_pc, OFFC, PERMC, INVC, A2C, NC);
  // out = 0.5*(aggW1 + aggC1 + bw1 + bc1) + hp @ WC1
  k_combine2<<<cdiv(P40, 256), 256, 0, stream>>>(A2W, A2C, bl1_w, bl1_c, out, NP, NC);
  gemm40(hp, WC1, out, NP, 1);
}